// GCNConv_15436112462151
// MI455X (gfx1250) — compile-verified
//
#include <hip/hip_runtime.h>
#include <hip/hip_bf16.h>

typedef float v2f __attribute__((ext_vector_type(2)));
typedef float v8f __attribute__((ext_vector_type(8)));

#define C_DIM 128

// ---------------------------------------------------------------------------
// Kernel 1: degree accumulation. deg[row[e]] += 1.0f
// ---------------------------------------------------------------------------
__global__ void gcn_degree_kernel(const int* __restrict__ rows,
                                  float* __restrict__ deg, int nedges) {
    int e = blockIdx.x * blockDim.x + threadIdx.x;
    if (e < nedges) {
        atomicAdd(&deg[rows[e]], 1.0f);
    }
}

// ---------------------------------------------------------------------------
// Kernel 2: inv_sqrt[n] = rsqrt(deg[n] + 1)
// ---------------------------------------------------------------------------
__global__ void gcn_invsqrt_kernel(const float* __restrict__ deg,
                                   float* __restrict__ inv_sqrt, int nnodes) {
    int n = blockIdx.x * blockDim.x + threadIdx.x;
    if (n < nnodes) {
        inv_sqrt[n] = rsqrtf(deg[n] + 1.0f);
    }
}

// ---------------------------------------------------------------------------
// Kernel 3: edge scatter. One wave32 per edge; lane handles 4 channels.
//   agg[row[e]][c] += x[col[e]][c] * inv_sqrt[col[e]]
// Edge id is uniform across the wave -> scalar loads; x gather and atomics
// are 512B contiguous per wave.
// ---------------------------------------------------------------------------
__global__ void gcn_scatter_kernel(const float* __restrict__ x,
                                   const int* __restrict__ rows,
                                   const int* __restrict__ cols,
                                   const float* __restrict__ inv_sqrt,
                                   float* __restrict__ agg, int nedges) {
    long long t = (long long)blockIdx.x * blockDim.x + threadIdx.x;
    int e = (int)(t >> 5);
    if (e >= nedges) return;
    int c = ((int)t & 31) * 4;

    int src = cols[e];
    int dst = rows[e];
    float s = inv_sqrt[src];

    const float4 xv = *(const float4*)(x + (size_t)src * C_DIM + c);
    float* p = agg + (size_t)dst * C_DIM + c;
    atomicAdd(p + 0, xv.x * s);
    atomicAdd(p + 1, xv.y * s);
    atomicAdd(p + 2, xv.z * s);
    atomicAdd(p + 3, xv.w * s);
}

// ---------------------------------------------------------------------------
// Kernel 4: finalize, in place:
//   agg[n][c] = (agg[n][c] + x[n][c]*inv_sqrt[n]) * inv_sqrt[n]
// One wave32 per node, float4 per lane.
// ---------------------------------------------------------------------------
__global__ void gcn_finalize_kernel(const float* __restrict__ x,
                                    const float* __restrict__ inv_sqrt,
                                    float* __restrict__ agg, int nnodes) {
    long long t = (long long)blockIdx.x * blockDim.x + threadIdx.x;
    int n = (int)(t >> 5);
    if (n >= nnodes) return;
    int c = ((int)t & 31) * 4;

    float s = inv_sqrt[n];
    const float4 xv = *(const float4*)(x + (size_t)n * C_DIM + c);
    float4* ap = (float4*)(agg + (size_t)n * C_DIM + c);
    float4 av = *ap;
    av.x = (av.x + xv.x * s) * s;
    av.y = (av.y + xv.y * s) * s;
    av.z = (av.z + xv.z * s) * s;
    av.w = (av.w + xv.w * s) * s;
    *ap = av;
}

// ---------------------------------------------------------------------------
// Kernel 5: out = H @ W^T using V_WMMA_F32_16X16X4_F32.
// H: [nnodes,128] f32, W: [128,128] f32 (row = out channel), out: [nnodes,128].
// Block = 256 threads = 8 waves. Block owns 16 output rows; wave w owns the
// 16-column tile [w*16, w*16+16).  K-loop: 32 x (16x16x4) WMMA.
//
// A-matrix (16x4 f32, 2 VGPRs): lane l -> M = l&15, K = (l>>4)*2 + vgpr.
// B-matrix (4x16 f32, 2 VGPRs): lane l -> N = l&15, K = (l>>4)*2 + vgpr.
//   B[k][n] = W^T[k][n] = W[n][k]  -> load W[colBase + (l&15)][k...]
// C/D (16x16 f32, 8 VGPRs): vgpr v, lane l -> M = v + (l>=16 ? 8 : 0), N = l&15.
// ---------------------------------------------------------------------------
__global__ void gcn_gemm_wmma_kernel(const float* __restrict__ H,
                                     const float* __restrict__ W,
                                     float* __restrict__ out, int nnodes) {
    const int lane = threadIdx.x & 31;
    const int wave = threadIdx.x >> 5;        // 0..7 -> column tile
    const int m = lane & 15;
    const int khalf = (lane >> 4) * 2;        // 0 or 2

    const int rowBase = blockIdx.x * 16;
    const int colBase = wave * 16;

    // Clamp row for loads so EXEC stays all-ones through the WMMA chain.
    int rowLd = rowBase + m;
    if (rowLd > nnodes - 1) rowLd = nnodes - 1;

    const float* hrow = H + (size_t)rowLd * C_DIM + khalf;
    const float* wrow = W + (size_t)(colBase + m) * C_DIM + khalf;

    v8f acc = {};
#pragma unroll
    for (int k0 = 0; k0 < C_DIM; k0 += 4) {
        v2f a = *(const v2f*)(hrow + k0);   // A[m][k0+khalf .. +1]
        v2f b = *(const v2f*)(wrow + k0);   // B[k0+khalf .. +1][m] = W[m][k..]
        acc = __builtin_amdgcn_wmma_f32_16x16x4_f32(
            /*neg_a=*/false, a, /*neg_b=*/false, b,
            /*c_mod=*/(short)0, acc, /*reuse_a=*/false, /*reuse_b=*/false);
    }

    const int rbase = rowBase + ((lane >> 4) * 8);
#pragma unroll
    for (int v = 0; v < 8; ++v) {
        int r = rbase + v;
        if (r < nnodes) {
            out[(size_t)r * C_DIM + colBase + m] = acc[v];
        }
    }
}

// ---------------------------------------------------------------------------
// Host launcher
// ---------------------------------------------------------------------------
extern "C" void kernel_launch(void* const* d_in, const int* in_sizes, int n_in,
                              void* d_out, int out_size, void* d_ws, size_t ws_size,
                              hipStream_t stream) {
    const float* x   = (const float*)d_in[0];                 // [N,128]
    const int*   ei  = (const int*)d_in[1];                   // [2,E] row-major
    const float* W   = (const float*)d_in[2];                 // [128,128]
    float*       out = (float*)d_out;                         // [N,128]

    const int nnodes = in_sizes[0] / C_DIM;
    const int nedges = in_sizes[1] / 2;
    const int* rows = ei;            // edge_index[0]
    const int* cols = ei + nedges;   // edge_index[1]

    // Workspace layout (256B aligned): deg[N] | inv_sqrt[N] | agg[N*128]
    auto align256 = [](size_t v) { return (v + 255) & ~(size_t)255; };
    char* ws = (char*)d_ws;
    size_t off = 0;
    float* deg = (float*)(ws + off);       off = align256(off + (size_t)nnodes * 4);
    float* inv_sqrt = (float*)(ws + off);  off = align256(off + (size_t)nnodes * 4);
    float* agg = (float*)(ws + off);       off = align256(off + (size_t)nnodes * C_DIM * 4);
    (void)ws_size;

    // Zero accumulators (graph-capture safe).
    hipMemsetAsync(deg, 0, (size_t)nnodes * 4, stream);
    hipMemsetAsync(agg, 0, (size_t)nnodes * C_DIM * 4, stream);

    // 1) degree
    {
        int threads = 256;
        int blocks = (nedges + threads - 1) / threads;
        gcn_degree_kernel<<<blocks, threads, 0, stream>>>(rows, deg, nedges);
    }
    // 2) inv_sqrt
    {
        int threads = 256;
        int blocks = (nnodes + threads - 1) / threads;
        gcn_invsqrt_kernel<<<blocks, threads, 0, stream>>>(deg, inv_sqrt, nnodes);
    }
    // 3) scatter-add over edges (1 wave per edge)
    {
        long long total = (long long)nedges * 32;
        int threads = 256;
        long long blocks = (total + threads - 1) / threads;
        gcn_scatter_kernel<<<(unsigned int)blocks, threads, 0, stream>>>(
            x, rows, cols, inv_sqrt, agg, nedges);
    }
    // 4) finalize (self contribution + right normalization), in place
    {
        long long total = (long long)nnodes * 32;
        int threads = 256;
        long long blocks = (total + threads - 1) / threads;
        gcn_finalize_kernel<<<(unsigned int)blocks, threads, 0, stream>>>(
            x, inv_sqrt, agg, nnodes);
    }
    // 5) WMMA GEMM: out = agg @ W^T
    {
        int threads = 256;  // 8 waves
        int blocks = (nnodes + 15) / 16;
        gcn_gemm_wmma_kernel<<<blocks, threads, 0, stream>>>(agg, W, out, nnodes);
    }
    (void)n_in; (void)out_size;
}